// SDFLoss_37641093382216
// MI455X (gfx1250) — compile-verified
//
#include <hip/hip_runtime.h>
#include <math.h>

typedef __attribute__((ext_vector_type(2))) float v2f;
typedef __attribute__((ext_vector_type(8))) float v8f;

// ---------------------------------------------------------------------------
// Kernel 1: se3 = exp_se3(pose_delta[b]) @ pose_init[b], one lane per pose.
// Also zeroes the loss accumulator (kernel 2 atomically adds into it).
// ---------------------------------------------------------------------------
__global__ __launch_bounds__(64)
void se3_pose_kernel(const float* __restrict__ pose_delta,
                     const float* __restrict__ pose_init,
                     float* out_loss, float* out_se3, int B)
{
    int b = blockIdx.x * blockDim.x + threadIdx.x;
    if (b == 0) *out_loss = 0.0f;
    if (b >= B) return;

    const float* d = pose_delta + 6 * b;
    float ux = d[0], uy = d[1], uz = d[2];
    float wx = d[3], wy = d[4], wz = d[5];

    float th2 = wx * wx + wy * wy + wz * wz;
    float th  = sqrtf(th2);
    bool  sm  = th < 1e-6f;
    float s   = sm ? 1.0f : th;
    float sn  = sinf(s);
    float cs  = cosf(s);
    float Ac  = sm ? (1.0f - th2 * (1.0f / 6.0f))          : (sn / s);
    float Bc  = sm ? (0.5f - th2 * (1.0f / 24.0f))         : ((1.0f - cs) / (s * s));
    float Cc  = sm ? (1.0f / 6.0f - th2 * (1.0f / 120.0f)) : ((s - sn) / (s * s * s));

    float K[3][3] = {{0.0f, -wz,  wy},
                     { wz, 0.0f, -wx},
                     {-wy,  wx, 0.0f}};
    float K2[3][3];
#pragma unroll
    for (int i = 0; i < 3; ++i)
#pragma unroll
        for (int j = 0; j < 3; ++j)
            K2[i][j] = K[i][0] * K[0][j] + K[i][1] * K[1][j] + K[i][2] * K[2][j];

    float R[3][3], V[3][3];
#pragma unroll
    for (int i = 0; i < 3; ++i)
#pragma unroll
        for (int j = 0; j < 3; ++j) {
            float I = (i == j) ? 1.0f : 0.0f;
            R[i][j] = I + Ac * K[i][j] + Bc * K2[i][j];
            V[i][j] = I + Bc * K[i][j] + Cc * K2[i][j];
        }
    float t[3];
#pragma unroll
    for (int i = 0; i < 3; ++i)
        t[i] = V[i][0] * ux + V[i][1] * uy + V[i][2] * uz;

    // se3 = [[R,t],[0,0,0,1]] @ pose_init   (pose_init bottom row is [0,0,0,1])
    const float* P = pose_init + 16 * b;
    float* O = out_se3 + 16 * b;
#pragma unroll
    for (int i = 0; i < 3; ++i)
#pragma unroll
        for (int j = 0; j < 4; ++j)
            O[i * 4 + j] = R[i][0] * P[j] + R[i][1] * P[4 + j] +
                           R[i][2] * P[8 + j] + t[i] * P[12 + j];
#pragma unroll
    for (int j = 0; j < 4; ++j) O[12 + j] = P[12 + j];
}

// ---------------------------------------------------------------------------
// Kernel 2: WMMA point transform (2 poses x 16 points per v_wmma_f32_16x16x4_f32,
// result lands one-point-per-lane) + trilinear SDF gather + loss reduction.
//   A(16x4): rows 0-2 = se3[2*by] rows [R|t], rows 8-10 = se3[2*by+1] rows.
//   B(4x16): column n = (x_n, y_n, z_n, 1).  Lane<16 holds K=0,1; lane>=16 K=2,3.
//   D: lane l<16 -> pose b point l; lane l>=16 -> pose b+1 point l-16; coords in d[0..2].
// ---------------------------------------------------------------------------
__global__ __launch_bounds__(256)
void sdf_sample_kernel(const float* se3,                    // [B][16] (in d_out)
                       const float* __restrict__ grid,      // [G][G][G]
                       const float* __restrict__ limits,    // [6]
                       const float* __restrict__ pts,       // [N][3]
                       float* sdf,                          // [B][N] (in d_out)
                       float* loss,
                       int B, int N, int G)
{
    const int lane  = threadIdx.x & 31;
    const int wid   = threadIdx.x >> 5;
    const int half  = lane >> 4;     // 0: K=0,1 / pose b ; 1: K=2,3 / pose b+1
    const int m     = lane & 15;
    const int poseA = 2 * (int)blockIdx.y;
    const int poseB = poseA + 1;

    // Build A-matrix fragment (2 VGPRs / lane), loop-invariant.
    v2f a = {0.0f, 0.0f};
    {
        int row = -1, pb = 0;
        if (m < 3)                { row = m;     pb = poseA; }
        else if (m >= 8 && m < 11){ row = m - 8; pb = poseB; }
        if (row >= 0 && pb < B) {
            const float* src = se3 + (size_t)pb * 16 + row * 4 + 2 * half;
            a.x = src[0];
            a.y = src[1];
        }
    }
    const int myPose = half ? poseB : poseA;

    // Grid mapping constants (uniform).
    const float lo0 = limits[0], lo1 = limits[1], lo2 = limits[2];
    const float hi0 = limits[3], hi1 = limits[4], hi2 = limits[5];
    const float gm1 = (float)(G - 1);
    const float sx = gm1 / (hi0 - lo0);
    const float sy = gm1 / (hi1 - lo1);
    const float sz = gm1 / (hi2 - lo2);
    const float uhi = gm1 - 1e-4f;   // clip upper bound => i0 <= G-2, i1 = i0+1 always

    const int tiles  = (N + 15) / 16;
    const int wavesG = gridDim.x * (blockDim.x >> 5);
    float acc = 0.0f;

    for (int t = (int)blockIdx.x * (blockDim.x >> 5) + wid; t < tiles; t += wavesG) {
        const int n  = t * 16 + m;
        const int nl = (n < N) ? n : (N - 1);

        // B-matrix fragment: point nl as (x,y,z,1) split across lane halves.
        const float* pp = pts + 3 * (size_t)nl;
        const float p0 = pp[0], p1 = pp[1], p2 = pp[2];
        v2f bv;
        bv.x = half ? p2 : p0;
        bv.y = half ? 1.0f : p1;

        v8f c = {0.f, 0.f, 0.f, 0.f, 0.f, 0.f, 0.f, 0.f};
        v8f dr = __builtin_amdgcn_wmma_f32_16x16x4_f32(
            false, a, false, bv, (short)0, c, false, false);

        const float px = dr[0], py = dr[1], pz = dr[2];

        // Trilinear sample (matches reference arithmetic).
        float u0 = fminf(fmaxf((px - lo0) * sx, 0.0f), uhi);
        float u1 = fminf(fmaxf((py - lo1) * sy, 0.0f), uhi);
        float u2 = fminf(fmaxf((pz - lo2) * sz, 0.0f), uhi);
        int ix0 = (int)u0, iy0 = (int)u1, iz0 = (int)u2;
        float fx = u0 - (float)ix0, fy = u1 - (float)iy0, fz = u2 - (float)iz0;
        int ix1 = ix0 + 1, iy1 = iy0 + 1;   // in-range by construction

        const int r00 = (ix0 * G + iy0) * G + iz0;
        const int r01 = (ix0 * G + iy1) * G + iz0;
        const int r10 = (ix1 * G + iy0) * G + iz0;
        const int r11 = (ix1 * G + iy1) * G + iz0;

        // z-pairs are contiguous: 4 x 8-byte gathers instead of 8 x 4-byte.
        float q00[2], q01[2], q10[2], q11[2];
        __builtin_memcpy(q00, grid + r00, 8);
        __builtin_memcpy(q01, grid + r01, 8);
        __builtin_memcpy(q10, grid + r10, 8);
        __builtin_memcpy(q11, grid + r11, 8);

        float c00 = q00[0] * (1.0f - fx) + q10[0] * fx;
        float c10 = q01[0] * (1.0f - fx) + q11[0] * fx;
        float c01 = q00[1] * (1.0f - fx) + q10[1] * fx;
        float c11 = q01[1] * (1.0f - fx) + q11[1] * fx;
        float c0  = c00 * (1.0f - fy) + c10 * fy;
        float c1  = c01 * (1.0f - fy) + c11 * fy;
        float sv  = c0 * (1.0f - fz) + c1 * fz;

        if (n < N && myPose < B) {
            sdf[(size_t)myPose * N + n] = sv;   // two contiguous 64B runs / wave
            acc += sv * sv;
        }
    }

    // Wave32 shuffle reduction, one atomic per wave.
#pragma unroll
    for (int off = 16; off > 0; off >>= 1)
        acc += __shfl_xor(acc, off, 32);
    if (lane == 0)
        atomicAdd(loss, 0.5f * acc);
    (void)wid;
}

// ---------------------------------------------------------------------------
extern "C" void kernel_launch(void* const* d_in, const int* in_sizes, int n_in,
                              void* d_out, int out_size, void* d_ws, size_t ws_size,
                              hipStream_t stream)
{
    const float* pose_delta = (const float*)d_in[0];
    const float* pose_init  = (const float*)d_in[1];
    const float* sdf_grids  = (const float*)d_in[2];
    const float* sdf_limits = (const float*)d_in[3];
    const float* points     = (const float*)d_in[4];

    const int B = in_sizes[0] / 6;
    const int N = in_sizes[4] / 3;
    long gsz = in_sizes[2];
    int G = 1;
    while ((long)G * G * G < gsz) ++G;

    float* out  = (float*)d_out;
    float* loss = out;                       // [0]
    float* sdf  = out + 1;                   // [1 .. B*N]
    float* se3  = out + 1 + (size_t)B * N;   // [1+B*N .. 1+B*N+16*B)

    se3_pose_kernel<<<(B + 63) / 64, 64, 0, stream>>>(pose_delta, pose_init,
                                                      loss, se3, B);

    const int tiles = (N + 15) / 16;
    int gx = (tiles + 63) / 64;              // ~8 tiles per wave (8 waves/block)
    if (gx < 1) gx = 1;
    dim3 grid(gx, (B + 1) / 2);
    sdf_sample_kernel<<<grid, 256, 0, stream>>>(se3, sdf_grids, sdf_limits, points,
                                                sdf, loss, B, N, G);
}